// SelectiveSSM_833223656443
// MI455X (gfx1250) — compile-verified
//
#include <hip/hip_runtime.h>
#include <math.h>

typedef __attribute__((ext_vector_type(2))) float v2f;
typedef __attribute__((ext_vector_type(8))) float v8f;

#define INNER_DIM 256
#define DT_SIZE   64
#define STATE_DIM 256
#define BATCH     2
#define SEQ_LEN   512
#define PROJ_N    (DT_SIZE + 2 * INNER_DIM)   /* 576 */
#define ROWS      (BATCH * SEQ_LEN)           /* 1024 */
#define CHUNKS    8
#define CHUNK_LEN (SEQ_LEN / CHUNKS)          /* 64  */

// ---------------------------------------------------------------------------
// One-wave 16x16 fp32 WMMA tile: C[16,16] = A[m0:+16, 0:K] @ B[0:K, n0:+16]
// V_WMMA_F32_16X16X4_F32 (wave32):
//   A 16x4 : lanes 0-15 -> M=lane, {K0,K1}; lanes 16-31 -> M=lane-16, {K2,K3}
//   B 4x16 : lanes 0-15 -> N=lane, {K0,K1}; lanes 16-31 -> N=lane-16, {K2,K3}
//   C/D    : VGPR r -> M = r + 8*(lane>=16), N = lane&15
// ---------------------------------------------------------------------------
__device__ __forceinline__ v8f wmma_tile_f32(const float* __restrict__ A, int lda,
                                             const float* __restrict__ B, int ldb,
                                             int m0, int n0, int K, int lane)
{
  const int half = lane >> 4;
  const int l16  = lane & 15;
  v8f acc = {0.f, 0.f, 0.f, 0.f, 0.f, 0.f, 0.f, 0.f};
  const float* arow = A + (size_t)(m0 + l16) * lda + half * 2;
  const float* bcol = B + (size_t)(half * 2) * ldb + (n0 + l16);
  for (int k = 0; k < K; k += 4) {
    v2f a, b;
    a.x = arow[k];
    a.y = arow[k + 1];
    b.x = bcol[(size_t)k * ldb];
    b.y = bcol[(size_t)(k + 1) * ldb];
    acc = __builtin_amdgcn_wmma_f32_16x16x4_f32(false, a, false, b,
                                                (short)0, acc, false, false);
  }
  return acc;
}

// ---------------------------------------------------------------------------
// Kernel 1: proj[1024,576] = x[1024,256] @ W_in[256,576]   (fp32 WMMA)
// ---------------------------------------------------------------------------
__global__ __launch_bounds__(128)
void proj_gemm_kernel(const float* __restrict__ X, const float* __restrict__ Win,
                      float* __restrict__ P)
{
  const int lane = threadIdx.x & 31;
  const int wave = threadIdx.x >> 5;
  const int tiles_n = PROJ_N / 16;                // 36
  const int tile = blockIdx.x * 4 + wave;         // 2304 tiles
  const int tm = tile / tiles_n;
  const int tn = tile - tm * tiles_n;
  const int m0 = tm << 4, n0 = tn << 4;

  v8f acc = wmma_tile_f32(X, INNER_DIM, Win, PROJ_N, m0, n0, INNER_DIM, lane);

  const int col  = n0 + (lane & 15);
  const int row0 = m0 + (lane >> 4) * 8;
#pragma unroll
  for (int r = 0; r < 8; ++r)
    P[(size_t)(row0 + r) * PROJ_N + col] = acc[r];
}

// ---------------------------------------------------------------------------
// Kernel 2: dt[1024,256] = softplus(proj[:, 0:64] @ W_dt[64,256] + b_dt)
// ---------------------------------------------------------------------------
__global__ __launch_bounds__(128)
void dt_gemm_softplus_kernel(const float* __restrict__ P, const float* __restrict__ Wdt,
                             const float* __restrict__ bdt, float* __restrict__ DT)
{
  const int lane = threadIdx.x & 31;
  const int wave = threadIdx.x >> 5;
  const int tiles_n = INNER_DIM / 16;             // 16
  const int tile = blockIdx.x * 4 + wave;         // 1024 tiles
  const int tm = tile / tiles_n;
  const int tn = tile - tm * tiles_n;
  const int m0 = tm << 4, n0 = tn << 4;

  v8f acc = wmma_tile_f32(P, PROJ_N, Wdt, INNER_DIM, m0, n0, DT_SIZE, lane);

  const int col  = n0 + (lane & 15);
  const int row0 = m0 + (lane >> 4) * 8;
  const float bias = bdt[col];
#pragma unroll
  for (int r = 0; r < 8; ++r) {
    float z  = acc[r] + bias;
    float sp = fmaxf(z, 0.f) + log1pf(__expf(-fabsf(z)));   // stable softplus
    DT[(size_t)(row0 + r) * INNER_DIM + col] = sp;
  }
}

// ---------------------------------------------------------------------------
// Kernel 3 (phase 1): per-chunk local scan from zero state.
// One wave32 per (b, d, chunk); lane owns 8 n-states in registers.
// Stores chunk end-state S_c and the scalar sum(dt) over the chunk
// (since prod_l exp(dt_l*alpha) == exp(alpha * sum_l dt_l)).
// ---------------------------------------------------------------------------
__global__ __launch_bounds__(256)
void scan_phase1_kernel(const float* __restrict__ X, const float* __restrict__ P,
                        const float* __restrict__ DT, const float* __restrict__ ALog,
                        float* __restrict__ Sc, float* __restrict__ DtSum)
{
  const int lane = threadIdx.x & 31;
  const int wid  = (blockIdx.x * blockDim.x + threadIdx.x) >> 5;  // 0..4095
  const int c  = wid & (CHUNKS - 1);
  const int bd = wid >> 3;            // b*256 + d, 0..511
  const int b  = bd >> 8;
  const int d  = bd & 255;
  const int n0 = lane * 8;

  float ac[8], s[8];
#pragma unroll
  for (int j = 0; j < 8; ++j) {
    ac[j] = -__expf(ALog[(size_t)d * STATE_DIM + n0 + j]);
    s[j]  = 0.f;
  }

  float dtsum = 0.f;
  const int l0 = c * CHUNK_LEN;
  for (int l = l0; l < l0 + CHUNK_LEN; ++l) {
    const size_t idx = (size_t)b * SEQ_LEN + l;
    const float dtv = DT[idx * INNER_DIM + d];
    const float xv  = X[idx * INNER_DIM + d];
    const float dtx = dtv * xv;
    dtsum += dtv;

    const float* row = P + idx * PROJ_N;
    const float4 be0 = *reinterpret_cast<const float4*>(row + DT_SIZE + n0);
    const float4 be1 = *reinterpret_cast<const float4*>(row + DT_SIZE + n0 + 4);
    const float be[8] = {be0.x, be0.y, be0.z, be0.w, be1.x, be1.y, be1.z, be1.w};
#pragma unroll
    for (int j = 0; j < 8; ++j)
      s[j] = fmaf(__expf(dtv * ac[j]), s[j], dtx * be[j]);
  }

  float* sc = Sc + ((size_t)bd * CHUNKS + c) * STATE_DIM + n0;
#pragma unroll
  for (int j = 0; j < 8; ++j) sc[j] = s[j];
  if (lane == 0) DtSum[(size_t)bd * CHUNKS + c] = dtsum;
}

// ---------------------------------------------------------------------------
// Kernel 4 (phase 2): stitch chunk carries. One thread per (b,d,n), 8 serial
// steps: carry <- exp(alpha*sum_dt_c)*carry + S_c ; S_c is rewritten in place
// with the chunk's INITIAL state.
// ---------------------------------------------------------------------------
__global__ __launch_bounds__(256)
void scan_phase2_kernel(const float* __restrict__ ALog, const float* __restrict__ DtSum,
                        float* __restrict__ Sc)
{
  const int flat = blockIdx.x * 256 + threadIdx.x;  // 0..131071
  const int bd = flat >> 8;                         // 0..511
  const int n  = flat & 255;
  const int d  = bd & 255;

  const float acn = -__expf(ALog[(size_t)d * STATE_DIM + n]);
  const float* dts = DtSum + (size_t)bd * CHUNKS;
  float* sc = Sc + (size_t)bd * CHUNKS * STATE_DIM + n;

  float carry = 0.f;
#pragma unroll
  for (int cc = 0; cc < CHUNKS; ++cc) {
    const float A   = __expf(acn * dts[cc]);
    const float tmp = sc[(size_t)cc * STATE_DIM];
    sc[(size_t)cc * STATE_DIM] = carry;             // chunk-initial state
    carry = fmaf(A, carry, tmp);
  }
}

// ---------------------------------------------------------------------------
// Kernel 5 (phase 3): re-run each chunk from its correct initial state,
// emitting y_t = sum_n(state*gamma) + x*delta. Wave32 butterfly reduce.
// ---------------------------------------------------------------------------
__global__ __launch_bounds__(256)
void scan_phase3_kernel(const float* __restrict__ X, const float* __restrict__ P,
                        const float* __restrict__ DT, const float* __restrict__ ALog,
                        const float* __restrict__ Delta, const float* __restrict__ Sc,
                        float* __restrict__ Y)
{
  const int lane = threadIdx.x & 31;
  const int wid  = (blockIdx.x * blockDim.x + threadIdx.x) >> 5;  // 0..4095
  const int c  = wid & (CHUNKS - 1);
  const int bd = wid >> 3;
  const int b  = bd >> 8;
  const int d  = bd & 255;
  const int n0 = lane * 8;

  float ac[8], s[8];
  const float* sc = Sc + ((size_t)bd * CHUNKS + c) * STATE_DIM + n0;
#pragma unroll
  for (int j = 0; j < 8; ++j) {
    ac[j] = -__expf(ALog[(size_t)d * STATE_DIM + n0 + j]);
    s[j]  = sc[j];                                  // chunk-initial state
  }
  const float deltad = Delta[d];

  const int l0 = c * CHUNK_LEN;
  for (int l = l0; l < l0 + CHUNK_LEN; ++l) {
    const size_t idx = (size_t)b * SEQ_LEN + l;
    const float dtv = DT[idx * INNER_DIM + d];
    const float xv  = X[idx * INNER_DIM + d];
    const float dtx = dtv * xv;

    const float* row = P + idx * PROJ_N;
    const float4 be0 = *reinterpret_cast<const float4*>(row + DT_SIZE + n0);
    const float4 be1 = *reinterpret_cast<const float4*>(row + DT_SIZE + n0 + 4);
    const float4 ga0 = *reinterpret_cast<const float4*>(row + DT_SIZE + INNER_DIM + n0);
    const float4 ga1 = *reinterpret_cast<const float4*>(row + DT_SIZE + INNER_DIM + n0 + 4);
    const float be[8] = {be0.x, be0.y, be0.z, be0.w, be1.x, be1.y, be1.z, be1.w};
    const float ga[8] = {ga0.x, ga0.y, ga0.z, ga0.w, ga1.x, ga1.y, ga1.z, ga1.w};

    float acc = 0.f;
#pragma unroll
    for (int j = 0; j < 8; ++j) {
      s[j] = fmaf(__expf(dtv * ac[j]), s[j], dtx * be[j]);
      acc  = fmaf(s[j], ga[j], acc);
    }
#pragma unroll
    for (int off = 16; off >= 1; off >>= 1)
      acc += __shfl_xor(acc, off, 32);

    if (lane == 0)
      Y[idx * INNER_DIM + d] = acc + xv * deltad;
  }
}

// ---------------------------------------------------------------------------
extern "C" void kernel_launch(void* const* d_in, const int* in_sizes, int n_in,
                              void* d_out, int out_size, void* d_ws, size_t ws_size,
                              hipStream_t stream)
{
  (void)in_sizes; (void)n_in; (void)out_size; (void)ws_size;

  const float* X     = (const float*)d_in[0];   // [B,L,D]
  const float* Win   = (const float*)d_in[1];   // [256,576]
  const float* Wdt   = (const float*)d_in[2];   // [64,256]
  const float* bdt   = (const float*)d_in[3];   // [256]
  const float* ALog  = (const float*)d_in[4];   // [256,256]
  const float* Delta = (const float*)d_in[5];   // [256]
  float* Y = (float*)d_out;

  float* P     = (float*)d_ws;                                   // 1024*576
  float* DT    = P     + (size_t)ROWS * PROJ_N;                  // 1024*256
  float* Sc    = DT    + (size_t)ROWS * INNER_DIM;               // 512*8*256
  float* DtSum = Sc    + (size_t)BATCH * INNER_DIM * CHUNKS * STATE_DIM;  // 512*8

  proj_gemm_kernel<<<(ROWS / 16) * (PROJ_N / 16) / 4, 128, 0, stream>>>(X, Win, P);
  dt_gemm_softplus_kernel<<<(ROWS / 16) * (INNER_DIM / 16) / 4, 128, 0, stream>>>(P, Wdt, bdt, DT);

  const int scan_waves  = BATCH * INNER_DIM * CHUNKS;            // 4096
  scan_phase1_kernel<<<scan_waves / 8, 256, 0, stream>>>(X, P, DT, ALog, Sc, DtSum);
  scan_phase2_kernel<<<(BATCH * INNER_DIM * STATE_DIM) / 256, 256, 0, stream>>>(ALog, DtSum, Sc);
  scan_phase3_kernel<<<scan_waves / 8, 256, 0, stream>>>(X, P, DT, ALog, Delta, Sc, Y);
}